// DirectVoxGO_86973087744212
// MI455X (gfx1250) — compile-verified
//
#include <hip/hip_runtime.h>

#define WAVE 32
#define WPB  8   // waves per 256-thread block

// ---------------------------------------------------------------------------
// Pass 0: zero per-ray [start,end) so empty rays read as count==0.
// ---------------------------------------------------------------------------
__global__ void init_bounds_kernel(int* __restrict__ start,
                                   int* __restrict__ end, int n_rays) {
    int i = blockIdx.x * blockDim.x + threadIdx.x;
    if (i < n_rays) { start[i] = 0; end[i] = 0; }
}

// ---------------------------------------------------------------------------
// Pass 1: ray_id is sorted; mark each segment boundary exactly once.
// ---------------------------------------------------------------------------
__global__ void find_bounds_kernel(const int* __restrict__ ray_id, int M,
                                   int* __restrict__ start,
                                   int* __restrict__ end) {
    int i = blockIdx.x * blockDim.x + threadIdx.x;
    if (i >= M) return;
    int r = ray_id[i];
    if (i == 0) start[r] = 0;
    if (i == M - 1) {
        end[r] = M;
    } else {
        int rn = ray_id[i + 1];
        if (rn != r) { end[r] = i + 1; start[rn] = i + 1; }
    }
}

// ---------------------------------------------------------------------------
// Pass 2: one wave32 per ray. Chunk k+1 streams HBM -> LDS via gfx1250
// global_load_async_to_lds_b32 (ASYNCcnt) while chunk k is composited.
// Segmented exclusive scan of log(1-alpha) via wave shuffles.
// ---------------------------------------------------------------------------
__global__ __launch_bounds__(256) void composite_kernel(
    const float* __restrict__ density, const float* __restrict__ rgb,
    const float* __restrict__ bgp, const float* __restrict__ shiftp,
    const float* __restrict__ intervalp, const int* __restrict__ start,
    const int* __restrict__ end, float* __restrict__ out, int n_rays) {

    __shared__ float sden[WPB][2][WAVE];       // density staging (double buffer)
    __shared__ float srgb[WPB][2][WAVE * 3];   // rgb staging

    const int lane = threadIdx.x & (WAVE - 1);
    const int wv   = threadIdx.x >> 5;
    const int ray  = blockIdx.x * WPB + wv;
    if (ray >= n_rays) return;

    const float shift    = *shiftp;
    const float interval = *intervalp;

    const int s = start[ray];
    const int e = end[ray];

    // Per-lane LDS byte offsets (low 32 bits of a generic LDS address are the
    // wave-relative LDS byte offset).
    const unsigned doff0 = (unsigned)(size_t)&sden[wv][0][lane];
    const unsigned doff1 = (unsigned)(size_t)&sden[wv][1][lane];
    const unsigned roff0 = (unsigned)(size_t)&srgb[wv][0][lane * 3];
    const unsigned roff1 = (unsigned)(size_t)&srgb[wv][1][lane * 3];

    // Issue 4 async HBM->LDS dword copies for chunk starting at `base` into
    // buffer `buf`. Caller guarantees at least one active lane (base < e).
    auto stage = [&](int buf, int base) {
        const int idx = base + lane;
        if (idx < e) {
            const unsigned d_off = buf ? doff1 : doff0;
            const unsigned r_off = buf ? roff1 : roff0;
            const unsigned long long ga =
                (unsigned long long)(size_t)(density + idx);
            const unsigned long long ra =
                (unsigned long long)(size_t)(rgb + 3 * (size_t)idx);
            asm volatile("global_load_async_to_lds_b32 %0, %1, off"
                         :: "v"(d_off), "v"(ga) : "memory");
            asm volatile("global_load_async_to_lds_b32 %0, %1, off"
                         :: "v"(r_off), "v"(ra) : "memory");
            asm volatile("global_load_async_to_lds_b32 %0, %1, off offset:4"
                         :: "v"(r_off), "v"(ra) : "memory");
            asm volatile("global_load_async_to_lds_b32 %0, %1, off offset:8"
                         :: "v"(r_off), "v"(ra) : "memory");
        }
    };

    float carry = 0.0f;                 // running log T at chunk start (uniform)
    float accR = 0.0f, accG = 0.0f, accB = 0.0f;

    int cur = 0;
    if (s < e) stage(0, s);             // prologue: stage first chunk

    for (int base = s; base < e; base += WAVE) {
        const int idx   = base + lane;
        const bool valid = (idx < e);
        const bool has_next = (base + WAVE < e);   // uniform across wave

        if (has_next) {
            stage(cur ^ 1, base + WAVE);           // overlap next chunk's DMA
            // Current chunk's 4 async ops complete in order before these 4.
            asm volatile("s_wait_asynccnt 0x4" ::: "memory");
        } else {
            asm volatile("s_wait_asynccnt 0x0" ::: "memory");
        }

        float lt = 0.0f, a = 0.0f, cr = 0.0f, cg = 0.0f, cb = 0.0f;
        if (valid) {
            // log(1-alpha) = -interval * softplus(density + shift), stable form.
            const float x  = sden[wv][cur][lane] + shift;
            const float sp = fmaxf(x, 0.0f) + log1pf(expf(-fabsf(x)));
            lt = -interval * sp;
            a  = -expm1f(lt);           // alpha
            cr = srgb[wv][cur][lane * 3 + 0];
            cg = srgb[wv][cur][lane * 3 + 1];
            cb = srgb[wv][cur][lane * 3 + 2];
        }

        // Wave32 inclusive scan of lt (Hillis-Steele, 5 shuffle steps).
        float incl = lt;
        #pragma unroll
        for (int d = 1; d < WAVE; d <<= 1) {
            float up = __shfl_up(incl, d, WAVE);
            if (lane >= d) incl += up;
        }
        const float excl = incl - lt;   // exclusive prefix within chunk

        // weight_i = alpha_i * exp(log T_i)
        const float w = a * expf(carry + excl);
        accR += w * cr;
        accG += w * cg;
        accB += w * cb;

        // Chunk total (invalid lanes contributed lt=0, so lane 31 holds it).
        carry += __shfl(incl, WAVE - 1, WAVE);
        cur ^= 1;
    }

    // Tree-reduce the per-lane partial sums.
    #pragma unroll
    for (int d = WAVE / 2; d > 0; d >>= 1) {
        accR += __shfl_down(accR, d, WAVE);
        accG += __shfl_down(accG, d, WAVE);
        accB += __shfl_down(accB, d, WAVE);
    }

    if (lane == 0) {
        const float T = expf(carry);    // final transmittance of the ray
        out[3 * (size_t)ray + 0] = accR + T * bgp[0];
        out[3 * (size_t)ray + 1] = accG + T * bgp[1];
        out[3 * (size_t)ray + 2] = accB + T * bgp[2];
    }
}

// ---------------------------------------------------------------------------
// Inputs (setup_inputs order):
//   0 density [M] f32, 1 rgb [M,3] f32, 2 bg [3] f32, 3 shift [1] f32,
//   4 interval [1] f32, 5 ray_id [M] i32, 6 n_rays [1] i32
// Output: [N,3] f32  (out_size == 3*N)
// ---------------------------------------------------------------------------
extern "C" void kernel_launch(void* const* d_in, const int* in_sizes, int n_in,
                              void* d_out, int out_size, void* d_ws, size_t ws_size,
                              hipStream_t stream) {
    const float* density  = (const float*)d_in[0];
    const float* rgb      = (const float*)d_in[1];
    const float* bg       = (const float*)d_in[2];
    const float* shift    = (const float*)d_in[3];
    const float* interval = (const float*)d_in[4];
    const int*   ray_id   = (const int*)d_in[5];

    const int M      = in_sizes[0];
    const int n_rays = out_size / 3;

    int* start = (int*)d_ws;
    int* end   = start + n_rays;

    float* out = (float*)d_out;

    {
        dim3 blk(256), grd((n_rays + 255) / 256);
        init_bounds_kernel<<<grd, blk, 0, stream>>>(start, end, n_rays);
    }
    {
        dim3 blk(256), grd((M + 255) / 256);
        find_bounds_kernel<<<grd, blk, 0, stream>>>(ray_id, M, start, end);
    }
    {
        dim3 blk(256), grd((n_rays + WPB - 1) / WPB);
        composite_kernel<<<grd, blk, 0, stream>>>(density, rgb, bg, shift,
                                                  interval, start, end, out,
                                                  n_rays);
    }
}